// torch_GRL_2465311228175
// MI455X (gfx1250) — compile-verified
//
#include <hip/hip_runtime.h>
#include <hip/hip_bf16.h>

typedef __attribute__((ext_vector_type(2))) float v2f;
typedef __attribute__((ext_vector_type(8))) float v8f;

#define NN 8192      // nodes
#define CF 128       // GCN feature width
#define RB 32        // rows per block in the big matmul
#define KC 128       // K-chunk staged in LDS
#define LSTRIDE 132  // padded LDS row stride (floats) -> conflict-free ds reads

// ---------------------------------------------------------------------------
// Pass 1: deg[i] = 1 + sum_j adj[i,j];  dinv[i] = rsqrt(deg[i])
// One block per row, coalesced float4 stream + block reduction.
// ---------------------------------------------------------------------------
__global__ __launch_bounds__(256) void deg_kernel(const float* __restrict__ adj,
                                                  float* __restrict__ dinv) {
    __shared__ float red[256];
    const int row = blockIdx.x;
    const float4* r4 = (const float4*)(adj + (size_t)row * NN);
    float s = 0.f;
    for (int i = threadIdx.x; i < NN / 4; i += 256) {
        float4 v = r4[i];
        s += v.x + v.y + v.z + v.w;
    }
    red[threadIdx.x] = s;
    __syncthreads();
    for (int off = 128; off > 0; off >>= 1) {
        if (threadIdx.x < off) red[threadIdx.x] += red[threadIdx.x + off];
        __syncthreads();
    }
    if (threadIdx.x == 0) dinv[row] = rsqrtf(red[0] + 1.0f);
}

// ---------------------------------------------------------------------------
// Generic tiny GEMM: out[n,c] = post( in1[n,:k1]·W[c,:k1] + in2[n,:k2]·W[c,k1:]
//                                     (+bias) ) with optional dinv / relu / mask.
// One thread per output element; W is tiny and L2-resident.
// ---------------------------------------------------------------------------
__global__ __launch_bounds__(256) void mlp_kernel(
    const float* __restrict__ in1, int k1,
    const float* __restrict__ in2, int k2,
    const float* __restrict__ W, const float* __restrict__ bias,
    const float* __restrict__ dinv, const float* __restrict__ mask,
    float* __restrict__ out, int cout, int relu) {
    const int idx = blockIdx.x * blockDim.x + threadIdx.x;
    const int n = idx / cout;
    const int c = idx % cout;
    if (n >= NN) return;
    const float* w = W + (size_t)c * (k1 + k2);
    float s = 0.f;
    const float* a1 = in1 + (size_t)n * k1;
    for (int k = 0; k < k1; ++k) s += a1[k] * w[k];
    if (in2) {
        const float* a2 = in2 + (size_t)n * k2;
        for (int k = 0; k < k2; ++k) s += a2[k] * w[k1 + k];
    }
    if (dinv) s *= dinv[n];
    if (bias) s += bias[c];
    if (relu) s = fmaxf(s, 0.f);
    if (mask) s *= mask[n];
    out[(size_t)n * cout + c] = s;
}

// ---------------------------------------------------------------------------
// Pass 2 (dominant): g = relu( dinv_i * ( (A + I) @ M )[i,c] + bg[c] )
// where M = dinv ⊙ (h @ wg^T) is precomputed.
// f32 WMMA 16x16x4. Block = 256 threads = 8 waves, owns a 32-row stripe.
// adj chunk (32 x 128) staged in LDS; each wave: 2 accumulator tiles
// (rows i0..15 and i0+16..31) x its 16-feature-column slice, B-frag shared.
// ---------------------------------------------------------------------------
__global__ __launch_bounds__(256) void agg_wmma_kernel(
    const float* __restrict__ adj, const float* __restrict__ M,
    const float* __restrict__ dinv, const float* __restrict__ bg,
    float* __restrict__ g) {
    __shared__ float lA[RB * LSTRIDE];

    const int i0   = blockIdx.x * RB;
    const int tid  = threadIdx.x;
    const int wave = tid >> 5;     // 0..7 -> feature column tile
    const int lane = tid & 31;
    const int n0   = wave * 16;
    const int half = lane >> 4;    // 0 | 1 (half-wave)
    const int l16  = lane & 15;

    v8f acc0 = {};
    v8f acc1 = {};

    // cooperative-load mapping: 8 threads/row, 16 consecutive floats each
    const int lrow = tid >> 3;            // 0..31
    const int lcol = (tid & 7) * 16;      // 0,16,...,112
    const float* gsrc = adj + (size_t)(i0 + lrow) * NN + lcol;

    for (int kc = 0; kc < NN; kc += KC) {
        // stage 32x128 adj chunk into LDS (coalesced 64B per thread)
        const float4* s4 = (const float4*)(gsrc + kc);
        float4 v0 = s4[0], v1 = s4[1], v2 = s4[2], v3 = s4[3];
        float4* d4 = (float4*)(lA + lrow * LSTRIDE + lcol);
        d4[0] = v0; d4[1] = v1; d4[2] = v2; d4[3] = v3;
        if (kc + KC < NN) __builtin_prefetch(gsrc + kc + KC, 0, 0);
        __syncthreads();

#pragma unroll 8
        for (int k = 0; k < KC; k += 4) {
            const int ka = k + half * 2;  // A/B frag K offset per half-wave
            v2f a0, a1, b;
            // A 16x4: lane l16 holds row l16; VGPR0/1 = K+{0,1} (half 0) or K+{2,3} (half 1)
            a0.x = lA[l16 * LSTRIDE + ka];
            a0.y = lA[l16 * LSTRIDE + ka + 1];
            a1.x = lA[(l16 + 16) * LSTRIDE + ka];
            a1.y = lA[(l16 + 16) * LSTRIDE + ka + 1];
            // B 4x16: VGPR r: lanes0-15 row K+r, lanes16-31 row K+r+2
            const float* mp = M + (size_t)(kc + ka) * CF + n0 + l16;
            b.x = mp[0];
            b.y = mp[CF];
            acc0 = __builtin_amdgcn_wmma_f32_16x16x4_f32(
                false, a0, false, b, (short)0, acc0, false, false);
            acc1 = __builtin_amdgcn_wmma_f32_16x16x4_f32(
                false, a1, false, b, (short)0, acc1, false, false);
        }
        __syncthreads();
    }

    // Epilogue. D layout: VGPR r -> (row r + 8*half, col l16)
    const int ncol = n0 + l16;
    const float bgc = bg[ncol];
#pragma unroll
    for (int r = 0; r < 8; ++r) {
        const int row0 = i0 + half * 8 + r;
        float v = acc0[r] + M[(size_t)row0 * CF + ncol];   // + self-loop
        v = dinv[row0] * v + bgc;
        g[(size_t)row0 * CF + ncol] = fmaxf(v, 0.f);
        const int row1 = row0 + 16;
        float w = acc1[r] + M[(size_t)row1 * CF + ncol];
        w = dinv[row1] * w + bgc;
        g[(size_t)row1 * CF + ncol] = fmaxf(w, 0.f);
    }
}

// ---------------------------------------------------------------------------
extern "C" void kernel_launch(void* const* d_in, const int* in_sizes, int n_in,
                              void* d_out, int out_size, void* d_ws, size_t ws_size,
                              hipStream_t stream) {
    const float* x    = (const float*)d_in[0];
    const float* adj  = (const float*)d_in[1];
    const float* mask = (const float*)d_in[2];
    const float* w1   = (const float*)d_in[3];
    const float* b1   = (const float*)d_in[4];
    const float* w2   = (const float*)d_in[5];
    const float* b2   = (const float*)d_in[6];
    const float* wg   = (const float*)d_in[7];
    const float* bg   = (const float*)d_in[8];
    const float* wd   = (const float*)d_in[9];
    const float* bd   = (const float*)d_in[10];
    const float* wp1  = (const float*)d_in[11];
    const float* bp1  = (const float*)d_in[12];
    const float* wp2  = (const float*)d_in[13];
    const float* bp2  = (const float*)d_in[14];
    const float* wo   = (const float*)d_in[15];
    const float* bo   = (const float*)d_in[16];

    float* ws   = (float*)d_ws;
    float* dinv = ws;                       // 8192
    float* h1   = dinv + NN;                // 8192*64
    float* h2   = h1 + (size_t)NN * 64;     // 8192*128
    float* Mv   = h2 + (size_t)NN * 128;    // 8192*128  (dinv ⊙ (h @ wg^T))
    float* gv   = Mv + (size_t)NN * 128;    // 8192*128
    float* g2   = gv + (size_t)NN * 128;    // 8192*128
    float* p1   = g2 + (size_t)NN * 128;    // 8192*128
    float* p2   = p1 + (size_t)NN * 128;    // 8192*64

    // degrees / dinv  (first full pass over adj)
    deg_kernel<<<NN, 256, 0, stream>>>(adj, dinv);

    // encoder MLP
    mlp_kernel<<<(NN * 64 + 255) / 256, 256, 0, stream>>>(
        x, 32, nullptr, 0, w1, b1, nullptr, nullptr, h1, 64, 1);
    mlp_kernel<<<(NN * 128 + 255) / 256, 256, 0, stream>>>(
        h1, 64, nullptr, 0, w2, b2, nullptr, nullptr, h2, 128, 1);

    // M = dinv ⊙ (h2 @ wg^T)   (no bias, no relu)
    mlp_kernel<<<(NN * 128 + 255) / 256, 256, 0, stream>>>(
        h2, 128, nullptr, 0, wg, nullptr, dinv, nullptr, Mv, 128, 0);

    // dominant op: g = relu(dinv ⊙ ((A+I) @ M) + bg)   — fp32 WMMA
    agg_wmma_kernel<<<NN / RB, 256, 0, stream>>>(adj, Mv, dinv, bg, gv);

    // decoder / head
    mlp_kernel<<<(NN * 128 + 255) / 256, 256, 0, stream>>>(
        gv, 128, nullptr, 0, wd, bd, nullptr, nullptr, g2, 128, 1);
    mlp_kernel<<<(NN * 128 + 255) / 256, 256, 0, stream>>>(
        g2, 128, h2, 128, wp1, bp1, nullptr, nullptr, p1, 128, 1);
    mlp_kernel<<<(NN * 64 + 255) / 256, 256, 0, stream>>>(
        p1, 128, nullptr, 0, wp2, bp2, nullptr, nullptr, p2, 64, 1);
    mlp_kernel<<<(NN * 8 + 255) / 256, 256, 0, stream>>>(
        p2, 64, nullptr, 0, wo, bo, nullptr, mask, (float*)d_out, 8, 0);
}